// L1_ChamferLoss_45337674776763
// MI455X (gfx1250) — compile-verified
//
#include <hip/hip_runtime.h>
#include <math.h>

typedef float v2f __attribute__((ext_vector_type(2)));
typedef float v8f __attribute__((ext_vector_type(8)));

#define NPTS 8192
#define TILES (NPTS / 16) // 512 column tiles / row tiles per batch

__global__ void chamfer_zero_kernel(float* out) {
    if (threadIdx.x == 0 && blockIdx.x == 0) out[0] = 0.0f;
}

// One wave per (direction, batch, 16-row tile). Streams all column tiles,
// computing 16x16 distance tiles with V_WMMA_F32_16X16X4_F32 and keeping a
// running elementwise min in the 8 accumulator VGPRs.
__global__ __launch_bounds__(256) void chamfer_wmma_kernel(
    const float* __restrict__ P1, const float* __restrict__ P2,
    float* __restrict__ out, int batches, float scale)
{
    const int waveInBlock = threadIdx.x >> 5;
    const int lane = threadIdx.x & 31;
    const int half = lane >> 4;   // 0: K0/K1 half, 1: K2/K3 half
    const int idx  = lane & 15;   // M (for A) / N (for B/C)

    int gw = blockIdx.x * 8 + waveInBlock;
    const int wavesPerDir = batches * TILES;
    const float* Asrc = P1;
    const float* Bsrc = P2;
    if (gw >= wavesPerDir) { gw -= wavesPerDir; Asrc = P2; Bsrc = P1; }
    const int batch   = gw / TILES;
    const int rowTile = gw - batch * TILES;

    // ---- A operand: row M = rowTile*16 + idx, A[M,:] = [-2ax,-2ay,-2az, 1] ----
    const float* ap = Asrc + ((size_t)batch * NPTS + (rowTile * 16 + idx)) * 3;
    const float ax = ap[0], ay = ap[1], az = ap[2];
    v2f av;
    av.x = half ? (-2.0f * az) : (-2.0f * ax);
    av.y = half ? 1.0f         : (-2.0f * ay);

    // Running min of D tiles, D[M,N] = -2 a.b + |b|^2
    v8f m;
#pragma unroll
    for (int r = 0; r < 8; ++r) m[r] = 3.0e38f;

    const float* bbase = Bsrc + (size_t)batch * NPTS * 3;
#pragma unroll 4
    for (int t = 0; t < TILES; ++t) {
        const float* bp = bbase + (size_t)(t * 16 + idx) * 3;
        const float bx = bp[0], by = bp[1], bz = bp[2];
        const float b2 = bx * bx + by * by + bz * bz;
        v2f bv;
        bv.x = half ? bz : bx;
        bv.y = half ? b2 : by;
        v8f c = {}; // C = 0 (|a|^2 folded in after the min-reduction)
        v8f d = __builtin_amdgcn_wmma_f32_16x16x4_f32(
            false, av, false, bv, (short)0, c, false, false);
#pragma unroll
        for (int r = 0; r < 8; ++r) m[r] = __builtin_fminf(m[r], d[r]);
    }

    // ---- reduce across the 16 lanes of each half: full row min ----
    float localSum = 0.0f;
#pragma unroll
    for (int r = 0; r < 8; ++r) {
        float v = m[r];
        v = __builtin_fminf(v, __shfl_xor(v, 1));
        v = __builtin_fminf(v, __shfl_xor(v, 2));
        v = __builtin_fminf(v, __shfl_xor(v, 4));
        v = __builtin_fminf(v, __shfl_xor(v, 8));
        if (idx == 0) {
            // add back |a|^2 for this row, clamp, sqrt
            const int row = rowTile * 16 + r + 8 * half;
            const float* qp = Asrc + ((size_t)batch * NPTS + row) * 3;
            const float qx = qp[0], qy = qp[1], qz = qp[2];
            float dd = qx * qx + qy * qy + qz * qz + v;
            dd = dd > 0.0f ? dd : 0.0f;
            localSum += sqrtf(dd);
        }
    }
    // combine half0 (lane 0) and half1 (lane 16) partial sums
    localSum += __shfl_xor(localSum, 16);
    if (lane == 0) atomicAdd(out, localSum * scale);
}

extern "C" void kernel_launch(void* const* d_in, const int* in_sizes, int n_in,
                              void* d_out, int out_size, void* d_ws, size_t ws_size,
                              hipStream_t stream) {
    (void)n_in; (void)out_size; (void)d_ws; (void)ws_size;
    const float* p1 = (const float*)d_in[0];
    const float* p2 = (const float*)d_in[1];
    float* out = (float*)d_out;

    const int batches = in_sizes[0] / (3 * NPTS); // 8
    // dist = (sum1 + sum2) / (2 * B * N)
    const float scale = 1.0f / (2.0f * (float)batches * (float)NPTS);

    chamfer_zero_kernel<<<1, 32, 0, stream>>>(out);

    const int totalWaves = 2 * batches * TILES;   // 8192 waves
    const int blocks = totalWaves / 8;            // 256 threads = 8 waves/block
    chamfer_wmma_kernel<<<blocks, 256, 0, stream>>>(p1, p2, out, batches, scale);
}